// PsRoIAlign_85272280694837
// MI455X (gfx1250) — compile-verified
//
#include <hip/hip_runtime.h>

typedef float v2f  __attribute__((ext_vector_type(2)));
typedef float v8f  __attribute__((ext_vector_type(8)));
typedef v2f   v2fu __attribute__((aligned(4)));   // 4B-aligned pair load

#define POOL   7
#define DCH    5
#define CCH    245      // DCH*POOL*POOL
#define HH     100
#define WW     100
#define NBOX   512
#define SSCALE 0.0625f

// PS-RoIAlign, wave32: lanes j / j+16 pair on one output (top/bottom bilinear
// rows).  Per sample each lane produces 2 weighted corner terms -> A[16x4] f32;
// 4 chained V_WMMA_F32_16X16X4_F32 with B == 1/(g*g) do the 16-term weighted
// reduction (row-sum) on the matrix pipe, accumulating in C.
// Corner pairs [xl, xl+1] are fetched with ONE unaligned b64 gather per sample
// (clamp edge handled by loading at min(xl, W-2) and selecting).
__global__ __launch_bounds__(256) void psroi_align_wmma(
    const float* __restrict__ x, const float* __restrict__ boxes,
    float* __restrict__ out, int total)
{
    const int lane = threadIdx.x & 31;
    const int wave = threadIdx.x >> 5;
    const int j    = lane & 15;     // output slot within the 16-output group
    const int half = lane >> 4;     // 0: y_low row terms, 1: y_high row terms
    const int gidx = blockIdx.x * 8 + wave;

    int o = gidx * 16 + j;
    if (o >= total) o = total - 1;          // clamp: keep EXEC all-ones for WMMA

    // flat output index o = ((r*DCH + d)*POOL + ph)*POOL + pw
    const int pw = o % POOL;
    int t = o / POOL;
    const int ph = t % POOL;
    t /= POOL;
    const int d  = t % DCH;
    const int r  = t / DCH;
    const int b  = r >> 9;                  // r / NBOX
    const int c  = (d * POOL + ph) * POOL + pw;   // position-sensitive channel

    const float4 box = ((const float4*)boxes)[r];
    const float start_w = box.x * SSCALE - 0.5f;
    const float start_h = box.y * SSCALE - 0.5f;
    const float bin_w = (box.z * SSCALE - 0.5f - start_w) * (1.0f / POOL);
    const float bin_h = (box.w * SSCALE - 0.5f - start_h) * (1.0f / POOL);

    const float* __restrict__ plane = x + (size_t)(b * CCH + c) * (HH * WW);

    v8f acc = {0.f, 0.f, 0.f, 0.f, 0.f, 0.f, 0.f, 0.f};
    v2f bm;  bm.x = 0.25f;  bm.y = 0.25f;   // B[k][n] = 1/(g*g)

    #pragma unroll
    for (int s = 0; s < 4; ++s) {           // g*g = 4 bilinear samples
        const int   iy = s >> 1, ix = s & 1;
        const float ys = start_h + ph * bin_h + ((float)iy + 0.5f) * 0.5f * bin_h;
        const float xs = start_w + pw * bin_w + ((float)ix + 0.5f) * 0.5f * bin_w;
        const float valid = ((ys >= -1.0f) && (ys <= (float)HH) &&
                             (xs >= -1.0f) && (xs <= (float)WW)) ? 1.0f : 0.0f;
        const float yc = fminf(fmaxf(ys, 0.0f), (float)(HH - 1));
        const float xc = fminf(fmaxf(xs, 0.0f), (float)(WW - 1));
        const int   yl = (int)floorf(yc);
        const int   xl = (int)floorf(xc);
        const int   yh = min(yl + 1, HH - 1);
        const float ly = yc - (float)yl;
        const float lx = xc - (float)xl;
        const float hy = 1.0f - ly, hx = 1.0f - lx;

        // low half: row y_low weighted hy; high half: row y_high weighted ly
        const int   row = half ? yh : yl;
        const float wy  = (half ? ly : hy) * valid;

        // one b64 gather fetches both corners of this row:
        //   pair = plane[row, xe .. xe+1],  xe = min(xl, W-2)  (always in-bounds)
        //   x[row, xl]   = (xl < W-1) ? pair.x : pair.y
        //   x[row, xh]   = pair.y            (xh = min(xl+1, W-1))
        const int  xe   = min(xl, WW - 2);
        const v2fu pair = *(const v2fu*)(plane + row * WW + xe);
        const float va  = (xl < WW - 1) ? pair.x : pair.y;
        const float vb  = pair.y;

        v2f a;                                       // A[j, 2k], A[j, 2k+1]
        a.x = wy * hx * va;
        a.y = wy * lx * vb;
        acc = __builtin_amdgcn_wmma_f32_16x16x4_f32(
                  false, a, false, bm, (short)0, acc, false, false);
    }

    // D[m,n] identical for every n; lane 0 holds outputs 0..7 (rows 0..7),
    // lane 16 holds outputs 8..15 (rows 8..15).  Two b128 stores per half.
    if (j == 0) {
        const int base = gidx * 16 + half * 8;
        if (base + 7 < total) {
            float* dst = out + (size_t)base;
            float4 lo = {acc[0], acc[1], acc[2], acc[3]};
            float4 hi = {acc[4], acc[5], acc[6], acc[7]};
            ((float4*)dst)[0] = lo;
            ((float4*)dst)[1] = hi;
        }
    }
}

extern "C" void kernel_launch(void* const* d_in, const int* in_sizes, int n_in,
                              void* d_out, int out_size, void* d_ws, size_t ws_size,
                              hipStream_t stream) {
    const float* x     = (const float*)d_in[0];   // [8,245,100,100] f32
    const float* boxes = (const float*)d_in[1];   // [8,512,4] f32
    float* out = (float*)d_out;                   // [4096,5,7,7] f32

    const int total  = out_size;                  // 1,003,520
    const int groups = (total + 15) / 16;         // 16 outputs per wave
    const int blocks = (groups + 7) / 8;          // 8 waves per block
    psroi_align_wmma<<<blocks, 256, 0, stream>>>(x, boxes, out, total);
}